// Transform_15616501088522
// MI455X (gfx1250) — compile-verified
//
#include <hip/hip_runtime.h>

// CDNA5 / gfx1250: wave32, WMMA 16x16x4 f32 path.
// Problem: Y = X @ W^T + b with X = (4194304 x 8) fp32, W 8x8, b 8.
// 256 MB traffic, 537 MFLOP -> HBM-bound at 23.3 TB/s (~11 us floor).
// Strategy: stream 16-row x 8-col tiles through the fp32 WMMA pipe,
// 2 tiles per wave-iteration, coalesced b128 stores, unconditional
// speculative prefetch one stride ahead.

typedef __attribute__((ext_vector_type(2))) float v2f;
typedef __attribute__((ext_vector_type(4))) float v4f;
typedef __attribute__((ext_vector_type(8))) float v8f;

__global__ __launch_bounds__(256) void block8x8_gemm_wmma(
    const float* __restrict__ x,     // [num_tiles*16][8]
    const float* __restrict__ w,     // [8][8]
    const float* __restrict__ bias,  // [8]
    float* __restrict__ y,           // [num_tiles*16][8]
    int num_tiles)
{
    const int lane  = threadIdx.x & 31;
    const int half  = lane >> 4;        // 0: lanes 0-15, 1: lanes 16-31
    const int m     = lane & 15;        // group-row within the 16-row tile
    const int wave  = blockIdx.x * (blockDim.x >> 5) + (threadIdx.x >> 5);
    const int nwav  = gridDim.x * (blockDim.x >> 5);

    // ---- A fragment: the 8x8 weight, rows 8..15 duplicate rows 0..7 ----
    // A(16x4) f32 layout (ISA 7.12.2): VGPR0 lanes0-15 K=0 / lanes16-31 K=2,
    //                                  VGPR1 lanes0-15 K=1 / lanes16-31 K=3.
    // A[row][K] = W[row&7][Kbase + K]  (so D holds Y^T: D-row = out column)
    const int wbase = (lane & 7) * 8 + 2 * half;
    v2f a0, a1;
    a0.x = w[wbase + 0];   // Kbase=0, K = 2*half
    a0.y = w[wbase + 1];   // Kbase=0, K = 2*half+1
    a1.x = w[wbase + 4];   // Kbase=4
    a1.y = w[wbase + 5];

    // ---- C init: D-row r corresponds to output column r (mod 8) ----
    // C VGPR r = bias[r], uniform across lanes (scalar loads).
    v8f cb;
    cb.s0 = bias[0]; cb.s1 = bias[1]; cb.s2 = bias[2]; cb.s3 = bias[3];
    cb.s4 = bias[4]; cb.s5 = bias[5]; cb.s6 = bias[6]; cb.s7 = bias[7];

    const int npairs = num_tiles >> 1;
    const int boff   = m * 8 + 2 * half;     // per-lane float2 offset in tile

    for (int p = wave; p < npairs; p += nwav) {
        const float* xt = x + (size_t)p * 256;   // two adjacent 512B tiles

        // Speculative streaming prefetch one wave-stride ahead; past-the-end
        // prefetches are silently dropped (no MEMVIOL for speculative TH).
        __builtin_prefetch(x + ((size_t)p + nwav) * 256, 0, 0);

        // ---- B fragments: X^T, B[K][n] = X[n][Kbase+K]; lanes0-15 take
        // K-half {0,1}, lanes16-31 {2,3} (same split as A). Four b64 loads.
        const v2f b0a = *(const v2f*)(xt + boff);            // tile0, Kbase=0
        const v2f b1a = *(const v2f*)(xt + boff + 4);        // tile0, Kbase=4
        const v2f b0b = *(const v2f*)(xt + 128 + boff);      // tile1, Kbase=0
        const v2f b1b = *(const v2f*)(xt + 128 + boff + 4);  // tile1, Kbase=4

        // Two independent accumulation chains keep the WMMA pipe busy.
        v8f da = __builtin_amdgcn_wmma_f32_16x16x4_f32(
            false, a0, false, b0a, (short)0, cb, false, false);
        v8f db = __builtin_amdgcn_wmma_f32_16x16x4_f32(
            false, a0, false, b0b, (short)0, cb, false, false);
        da = __builtin_amdgcn_wmma_f32_16x16x4_f32(
            false, a1, false, b1a, (short)0, da, false, false);
        db = __builtin_amdgcn_wmma_f32_16x16x4_f32(
            false, a1, false, b1b, (short)0, db, false, false);

        // ---- Store: lane owns output row m, cols 4*half..4*half+3 (upper
        // D rows duplicate lower ones), so each tile is one fully-active
        // contiguous 512B global_store_b128.
        float* yt = y + (size_t)p * 256 + boff - 2 * half + 4 * half; // = m*8+4*half
        v4f oa, ob;
        oa.x = half ? da.s4 : da.s0;
        oa.y = half ? da.s5 : da.s1;
        oa.z = half ? da.s6 : da.s2;
        oa.w = half ? da.s7 : da.s3;
        ob.x = half ? db.s4 : db.s0;
        ob.y = half ? db.s5 : db.s1;
        ob.z = half ? db.s6 : db.s2;
        ob.w = half ? db.s7 : db.s3;
        *(v4f*)(yt)       = oa;
        *(v4f*)(yt + 128) = ob;
    }

    // Odd-tile tail (not hit for the reference shape; wave-uniform guard
    // keeps EXEC all-ones through the WMMAs).
    if ((num_tiles & 1) && wave == 0) {
        const int t = num_tiles - 1;
        const float* xt = x + (size_t)t * 128;
        const v2f b0 = *(const v2f*)(xt + boff);
        const v2f b1 = *(const v2f*)(xt + boff + 4);
        v8f d = __builtin_amdgcn_wmma_f32_16x16x4_f32(
            false, a0, false, b0, (short)0, cb, false, false);
        d = __builtin_amdgcn_wmma_f32_16x16x4_f32(
            false, a1, false, b1, (short)0, d, false, false);
        v4f out;
        out.x = half ? d.s4 : d.s0;
        out.y = half ? d.s5 : d.s1;
        out.z = half ? d.s6 : d.s2;
        out.w = half ? d.s7 : d.s3;
        *(v4f*)(y + (size_t)t * 128 + m * 8 + 4 * half) = out;
    }
}

extern "C" void kernel_launch(void* const* d_in, const int* in_sizes, int n_in,
                              void* d_out, int out_size, void* d_ws, size_t ws_size,
                              hipStream_t stream) {
    const float* x    = (const float*)d_in[0];   // [256,128,32,32] fp32
    const float* wmat = (const float*)d_in[1];   // [8,8] fp32
    const float* bias = (const float*)d_in[2];   // [8] fp32
    float* y          = (float*)d_out;

    const int num_tiles = in_sizes[0] / 128;     // 16 groups of 8 per tile
    const int npairs    = num_tiles >> 1;

    // 1024 blocks x 8 waves = 8192 waves -> ~16 tile-pairs per wave.
    int blocks = 1024;
    int max_blocks = (npairs + 7) / 8;
    if (blocks > max_blocks) blocks = max_blocks;
    if (blocks < 1) blocks = 1;

    block8x8_gemm_wmma<<<blocks, 256, 0, stream>>>(x, wmat, bias, y, num_tiles);
}